// MPN_2379411882636
// MI455X (gfx1250) — compile-verified
//
#include <hip/hip_runtime.h>

#define HIDDEN   128
#define AFD      39          // ATOM_FDIM
#define IN_FDIM  50          // ATOM_FDIM + BOND_FDIM
#define MAX_NB   6

typedef __attribute__((ext_vector_type(2))) float v2f;
typedef __attribute__((ext_vector_type(8))) float v8f;

// D = A(16x4) * B(4x16) + C, fp32 WMMA (ISA: V_WMMA_F32_16X16X4_F32)
__device__ __forceinline__ v8f wmma4(v2f a, v2f b, v8f c) {
  return __builtin_amdgcn_wmma_f32_16x16x4_f32(false, a, false, b, (short)0, c,
                                               false, false);
}

__device__ __forceinline__ v2f ld2(const float* p) { return *(const v2f*)p; }

// ---------------------------------------------------------------------------
// Kernel 1: binput = fbonds @ W_i ; msg = relu(binput)
// Tile: 64 bonds x 128 out, K = 50 padded to 52.
// W_i kept transposed in LDS: sBt[N=128][K pad 54] -> B frag = one b64 load.
// ---------------------------------------------------------------------------
__global__ __launch_bounds__(256)
void k_bond_input(const float* __restrict__ fbonds, const float* __restrict__ W_i,
                  float* __restrict__ binput, float* __restrict__ msg, int n_bonds)
{
  __shared__ __align__(16) float sA[64][52];        // fbonds tile, cols>=50 zero
  __shared__ __align__(16) float sBt[HIDDEN][54];   // W_i^T, k>=50 zero
  const int tid = threadIdx.x;
  const int m0  = blockIdx.x * 64;

  // W_i rows (k) read coalesced as float4 along N, stored transposed
  for (int i = tid; i < 54 * 32; i += 256) {
    const int k  = i >> 5;        // 0..53
    const int cg = i & 31;        // N group of 4
    float4 w = make_float4(0.f, 0.f, 0.f, 0.f);
    if (k < IN_FDIM) w = *(const float4*)(W_i + (size_t)k * HIDDEN + cg * 4);
    sBt[cg * 4 + 0][k] = w.x;
    sBt[cg * 4 + 1][k] = w.y;
    sBt[cg * 4 + 2][k] = w.z;
    sBt[cg * 4 + 3][k] = w.w;
  }
  for (int i = tid; i < 64 * 52; i += 256) {
    const int r = i / 52, c = i % 52;
    const int row = m0 + r;
    sA[r][c] = (c < IN_FDIM && row < n_bonds)
                 ? fbonds[(size_t)row * IN_FDIM + c] : 0.f;
  }
  __syncthreads();

  const int lane = tid & 31;
  const int hf   = lane >> 4;        // half-wave
  const int l    = lane & 15;
  const int wv   = tid >> 5;
  const int mt   = wv >> 1;          // 0..3 : 16-row subtile
  const int nb4  = (wv & 1) * 4;     // base 16-col subtile (0 or 4)

  v8f acc[4];
#pragma unroll
  for (int t = 0; t < 4; ++t)
#pragma unroll
    for (int e = 0; e < 8; ++e) acc[t][e] = 0.f;

#pragma unroll
  for (int kb = 0; kb < 13; ++kb) {  // K = 52
    const int kk = kb * 4 + 2 * hf;  // even -> 8B aligned
    const v2f a = ld2(&sA[mt * 16 + l][kk]);
#pragma unroll
    for (int t = 0; t < 4; ++t) {
      const v2f b = ld2(&sBt[(nb4 + t) * 16 + l][kk]);
      acc[t] = wmma4(a, b, acc[t]);
    }
  }

#pragma unroll
  for (int t = 0; t < 4; ++t) {
    const int col = (nb4 + t) * 16 + l;
#pragma unroll
    for (int r = 0; r < 8; ++r) {
      const int row = m0 + mt * 16 + r + 8 * hf;
      if (row < n_bonds) {
        const float v = acc[t][r];
        binput[(size_t)row * HIDDEN + col] = v;
        msg[(size_t)row * HIDDEN + col]    = v > 0.f ? v : 0.f;
      }
    }
  }
}

// ---------------------------------------------------------------------------
// Kernel 2: nei = gather-sum(msg_in, bgraph); msg_out = relu(binput + nei@W_h)
// W_h streamed in transposed K-chunks: sWt[N=128][K pad 34].
// ---------------------------------------------------------------------------
__global__ __launch_bounds__(256)
void k_depth(const float* __restrict__ msg_in, const int* __restrict__ bgraph,
             const float* __restrict__ W_h, const float* __restrict__ binput,
             float* __restrict__ msg_out, int n_bonds)
{
  __shared__ __align__(16) float sN[64][132];       // gathered nei tile
  __shared__ __align__(16) float sWt[HIDDEN][34];   // W_h chunk, transposed
  const int tid = threadIdx.x;
  const int m0  = blockIdx.x * 64;

  // gather-sum of 6 neighbor bond messages, float4 per task
  for (int task = tid; task < 64 * 32; task += 256) {
    const int bl = task >> 5;
    const int fg = task & 31;
    const int bond = m0 + bl;
    float4 s = make_float4(0.f, 0.f, 0.f, 0.f);
    if (bond < n_bonds) {
#pragma unroll
      for (int j = 0; j < MAX_NB; ++j) {
        const int nb = bgraph[bond * MAX_NB + j];
        const float4 m = *(const float4*)(msg_in + (size_t)nb * HIDDEN + fg * 4);
        s.x += m.x; s.y += m.y; s.z += m.z; s.w += m.w;
      }
    }
    *(float4*)&sN[bl][fg * 4] = s;
  }

  const int lane = tid & 31;
  const int hf   = lane >> 4;
  const int l    = lane & 15;
  const int wv   = tid >> 5;
  const int mt   = wv >> 1;
  const int nb4  = (wv & 1) * 4;

  v8f acc[4];
#pragma unroll
  for (int t = 0; t < 4; ++t)
#pragma unroll
    for (int e = 0; e < 8; ++e) acc[t][e] = 0.f;

  for (int kp = 0; kp < 4; ++kp) {            // K chunks of 32 (K=128)
    __syncthreads();
    for (int i = tid; i < 32 * 32; i += 256) {
      const int k  = i >> 5;                  // k within chunk
      const int cg = i & 31;                  // N group of 4
      const float4 w =
          *(const float4*)(W_h + (size_t)(kp * 32 + k) * HIDDEN + cg * 4);
      sWt[cg * 4 + 0][k] = w.x;
      sWt[cg * 4 + 1][k] = w.y;
      sWt[cg * 4 + 2][k] = w.z;
      sWt[cg * 4 + 3][k] = w.w;
    }
    __syncthreads();
#pragma unroll
    for (int kb = 0; kb < 8; ++kb) {
      const int kk = kb * 4 + 2 * hf;
      const v2f a = ld2(&sN[mt * 16 + l][kp * 32 + kk]);
#pragma unroll
      for (int t = 0; t < 4; ++t) {
        const v2f b = ld2(&sWt[(nb4 + t) * 16 + l][kk]);
        acc[t] = wmma4(a, b, acc[t]);
      }
    }
  }

#pragma unroll
  for (int t = 0; t < 4; ++t) {
    const int col = (nb4 + t) * 16 + l;
#pragma unroll
    for (int r = 0; r < 8; ++r) {
      const int row = m0 + mt * 16 + r + 8 * hf;
      if (row < n_bonds) {
        float v = acc[t][r] + binput[(size_t)row * HIDDEN + col];
        msg_out[(size_t)row * HIDDEN + col] = v > 0.f ? v : 0.f;
      }
    }
  }
}

// ---------------------------------------------------------------------------
// Kernel 3: nei = gather-sum(msg, agraph);
//           h = relu([fatoms|nei] @ W_o + b_o); atomic segment-sum into sums.
// ---------------------------------------------------------------------------
__global__ __launch_bounds__(256)
void k_atom(const float* __restrict__ msg, const int* __restrict__ agraph,
            const float* __restrict__ fatoms, const float* __restrict__ W_o,
            const float* __restrict__ b_o, const int* __restrict__ mol_ids,
            float* __restrict__ mol_sums, int n_atoms)
{
  __shared__ __align__(16) float sN[64][132];       // gathered nei
  __shared__ __align__(16) float sF[64][50];        // fatoms tile (c>=39 zero)
  __shared__ __align__(16) float sWt[HIDDEN][34];   // W_o chunk, transposed
  const int tid = threadIdx.x;
  const int m0  = blockIdx.x * 64;

  for (int task = tid; task < 64 * 32; task += 256) {
    const int al = task >> 5;
    const int fg = task & 31;
    const int atom = m0 + al;
    float4 s = make_float4(0.f, 0.f, 0.f, 0.f);
    if (atom < n_atoms) {
#pragma unroll
      for (int j = 0; j < MAX_NB; ++j) {
        const int nb = agraph[atom * MAX_NB + j];
        const float4 m = *(const float4*)(msg + (size_t)nb * HIDDEN + fg * 4);
        s.x += m.x; s.y += m.y; s.z += m.z; s.w += m.w;
      }
    }
    *(float4*)&sN[al][fg * 4] = s;
  }
  for (int i = tid; i < 64 * 50; i += 256) {
    const int r = i / 50, c = i % 50;
    const int row = m0 + r;
    sF[r][c] = (c < AFD && row < n_atoms) ? fatoms[(size_t)row * AFD + c] : 0.f;
  }

  const int lane = tid & 31;
  const int hf   = lane >> 4;
  const int l    = lane & 15;
  const int wv   = tid >> 5;
  const int mt   = wv >> 1;
  const int nb4  = (wv & 1) * 4;

  v8f acc[4];
#pragma unroll
  for (int t = 0; t < 4; ++t)
#pragma unroll
    for (int e = 0; e < 8; ++e) acc[t][e] = 0.f;

  // Stage 1: fatoms part, K = 39 padded to 48, chunks of 16
  for (int kp = 0; kp < 3; ++kp) {
    __syncthreads();
    for (int i = tid; i < 16 * 32; i += 256) {
      const int k  = i >> 5;
      const int cg = i & 31;
      const int gr = kp * 16 + k;
      float4 w = make_float4(0.f, 0.f, 0.f, 0.f);
      if (gr < AFD) w = *(const float4*)(W_o + (size_t)gr * HIDDEN + cg * 4);
      sWt[cg * 4 + 0][k] = w.x;
      sWt[cg * 4 + 1][k] = w.y;
      sWt[cg * 4 + 2][k] = w.z;
      sWt[cg * 4 + 3][k] = w.w;
    }
    __syncthreads();
#pragma unroll
    for (int kb = 0; kb < 4; ++kb) {
      const int kk = kb * 4 + 2 * hf;
      const v2f a = ld2(&sF[mt * 16 + l][kp * 16 + kk]);
#pragma unroll
      for (int t = 0; t < 4; ++t) {
        const v2f b = ld2(&sWt[(nb4 + t) * 16 + l][kk]);
        acc[t] = wmma4(a, b, acc[t]);
      }
    }
  }
  // Stage 2: nei part, K = 128, chunks of 32, B rows = W_o[39 + k]
  for (int kp = 0; kp < 4; ++kp) {
    __syncthreads();
    for (int i = tid; i < 32 * 32; i += 256) {
      const int k  = i >> 5;
      const int cg = i & 31;
      const float4 w =
          *(const float4*)(W_o + (size_t)(AFD + kp * 32 + k) * HIDDEN + cg * 4);
      sWt[cg * 4 + 0][k] = w.x;
      sWt[cg * 4 + 1][k] = w.y;
      sWt[cg * 4 + 2][k] = w.z;
      sWt[cg * 4 + 3][k] = w.w;
    }
    __syncthreads();
#pragma unroll
    for (int kb = 0; kb < 8; ++kb) {
      const int kk = kb * 4 + 2 * hf;
      const v2f a = ld2(&sN[mt * 16 + l][kp * 32 + kk]);
#pragma unroll
      for (int t = 0; t < 4; ++t) {
        const v2f b = ld2(&sWt[(nb4 + t) * 16 + l][kk]);
        acc[t] = wmma4(a, b, acc[t]);
      }
    }
  }

#pragma unroll
  for (int t = 0; t < 4; ++t) {
    const int col = (nb4 + t) * 16 + l;
    const float bo = b_o[col];
#pragma unroll
    for (int r = 0; r < 8; ++r) {
      const int row = m0 + mt * 16 + r + 8 * hf;
      if (row < n_atoms) {
        float v = acc[t][r] + bo;
        v = v > 0.f ? v : 0.f;
        atomicAdd(mol_sums + (size_t)mol_ids[row] * HIDDEN + col, v);
      }
    }
  }
}

// ---------------------------------------------------------------------------
__global__ void k_zero(float* __restrict__ p, int n) {
  const int i = blockIdx.x * blockDim.x + threadIdx.x;
  if (i < n) p[i] = 0.f;
}

__global__ void k_counts(const int* __restrict__ mol_ids,
                         float* __restrict__ counts, int n_atoms) {
  const int i = blockIdx.x * blockDim.x + threadIdx.x;
  if (i < n_atoms) atomicAdd(&counts[mol_ids[i]], 1.f);
}

__global__ void k_final(const float* __restrict__ sums,
                        const float* __restrict__ counts,
                        float* __restrict__ out, int n) {
  const int i = blockIdx.x * blockDim.x + threadIdx.x;
  if (i < n) {
    const float c = counts[i >> 7];
    out[i] = sums[i] / fmaxf(c, 1.f);
  }
}

// ---------------------------------------------------------------------------
extern "C" void kernel_launch(void* const* d_in, const int* in_sizes, int n_in,
                              void* d_out, int out_size, void* d_ws, size_t ws_size,
                              hipStream_t stream)
{
  const float* fatoms  = (const float*)d_in[0];
  const float* fbonds  = (const float*)d_in[1];
  const int*   agraph  = (const int*)d_in[2];
  const int*   bgraph  = (const int*)d_in[3];
  const int*   mol_ids = (const int*)d_in[4];
  const float* W_i     = (const float*)d_in[5];
  const float* W_h     = (const float*)d_in[6];
  const float* W_o     = (const float*)d_in[7];
  const float* b_o     = (const float*)d_in[8];

  const int n_atoms = in_sizes[0] / AFD;
  const int n_bonds = in_sizes[1] / IN_FDIM;
  const int n_mols  = out_size / HIDDEN;

  float* ws     = (float*)d_ws;
  float* binput = ws;
  float* msgA   = binput + (size_t)n_bonds * HIDDEN;
  float* msgB   = msgA   + (size_t)n_bonds * HIDDEN;
  float* sums   = msgB   + (size_t)n_bonds * HIDDEN;
  float* counts = sums   + (size_t)n_mols * HIDDEN;   // contiguous after sums

  float* out = (float*)d_out;

  {
    const int n = n_mols * HIDDEN + n_mols;           // zero sums + counts
    k_zero<<<(n + 255) / 256, 256, 0, stream>>>(sums, n);
  }

  const int gb = (n_bonds + 63) / 64;
  k_bond_input<<<gb, 256, 0, stream>>>(fbonds, W_i, binput, msgA, n_bonds);

  float* cur = msgA;
  float* nxt = msgB;
  for (int d = 0; d < 4; ++d) {                       // DEPTH-1 iterations
    k_depth<<<gb, 256, 0, stream>>>(cur, bgraph, W_h, binput, nxt, n_bonds);
    float* t = cur; cur = nxt; nxt = t;
  }

  const int ga = (n_atoms + 63) / 64;
  k_atom<<<ga, 256, 0, stream>>>(cur, agraph, fatoms, W_o, b_o, mol_ids,
                                 sums, n_atoms);
  k_counts<<<(n_atoms + 255) / 256, 256, 0, stream>>>(mol_ids, counts, n_atoms);
  k_final<<<(n_mols * HIDDEN + 255) / 256, 256, 0, stream>>>(sums, counts, out,
                                                             n_mols * HIDDEN);
}